// GraphNeuralNetwork_56049323213740
// MI455X (gfx1250) — compile-verified
//
#include <hip/hip_runtime.h>
#include <math.h>
#include <stdint.h>

typedef __attribute__((ext_vector_type(2))) float v2f;
typedef __attribute__((ext_vector_type(8))) float v8f;
typedef int vi4 __attribute__((vector_size(4 * sizeof(int))));

// ---- native L2 f32 atomic add (no return -> STOREcnt, retired at L2) -------
__device__ __forceinline__ void atomAddF(float* p, float v) {
    asm volatile("global_atomic_add_f32 %0, %1, off" :: "v"(p), "v"(v) : "memory");
}

// ---- gfx1250 async global->LDS b128 copy, with graceful fallback -----------
__device__ __forceinline__ void asyncCopy16(const float* g, float* l) {
#if __has_builtin(__builtin_amdgcn_global_load_async_to_lds_b128)
    __builtin_amdgcn_global_load_async_to_lds_b128(
        (__attribute__((address_space(1))) vi4*)g,
        (__attribute__((address_space(3))) vi4*)l, 0, 0);
#else
    *(float4*)l = *(const float4*)g;
#endif
}
__device__ __forceinline__ void waitAsync() {
#if __has_builtin(__builtin_amdgcn_global_load_async_to_lds_b128)
  #if __has_builtin(__builtin_amdgcn_s_wait_asynccnt)
    __builtin_amdgcn_s_wait_asynccnt(0);
  #else
    asm volatile("s_wait_asynccnt 0x0" ::: "memory");
  #endif
#endif
}

// ---------------------------------------------------------------- degree / norm
__global__ __launch_bounds__(256) void k_degree(const int* __restrict__ dst,
                                                float* __restrict__ deg, int E) {
    int e = blockIdx.x * 256 + threadIdx.x;
    if (e < E) atomAddF(&deg[dst[e]], 1.0f);
}

__global__ __launch_bounds__(256) void k_rsqrt(float* __restrict__ d, int Nn) {
    int i = blockIdx.x * 256 + threadIdx.x;
    if (i < Nn) d[i] = rsqrtf(d[i] + 1.0f);   // +1 = self loop; deg >= 1 guaranteed
}

// ---------------------------------------------------------------- WMMA f32 GEMM
// C[M x Nc] = act(A[M x K]) @ B[K x Nc], act = relu if reluIn.
// Block: 256 threads = 8 waves as 4 row-groups x 2 col-groups.
// Wave tile 16x32 (two v8f accumulators share one A operand). Block tile 64x64.
// A staged synchronously (fused relu + M-guard); B staged with async b128 copies.
__global__ __launch_bounds__(256) void k_gemm_wmma(const float* __restrict__ A,
                                                   const float* __restrict__ B,
                                                   float* __restrict__ C,
                                                   int M, int K, int Nc, int reluIn) {
    __shared__ float As[64][36];   // [row][k]  pad->36: rows 144B apart, 16B aligned
    __shared__ float Bs[32][68];   // [k][col]  pad->68: rows 272B apart, 16B aligned

    const int tid     = threadIdx.x;
    const int rowBase = blockIdx.x * 64;
    const int colBase = blockIdx.y * 64;
    const int waveId  = tid >> 5;
    const int lane    = tid & 31;
    const int half    = lane >> 4;            // 0: lanes 0-15, 1: lanes 16-31
    const int lm      = lane & 15;
    const int wr      = (waveId >> 1) * 16;   // 0/16/32/48
    const int wc      = (waveId & 1) * 32;    // 0/32

    v8f acc0 = {};
    v8f acc1 = {};

    for (int k0 = 0; k0 < K; k0 += 32) {
        // ---- B: 32k x 64col chunk, 512 float4 -> 2 async copies per thread
#pragma unroll
        for (int it = 0; it < 2; ++it) {
            int idx = tid + it * 256;
            int r   = idx >> 4;               // k row in chunk
            int c4  = (idx & 15) << 2;        // col in chunk
            asyncCopy16(&B[(size_t)(k0 + r) * Nc + (colBase + c4)], &Bs[r][c4]);
        }
        // ---- A: 64row x 32k chunk, 512 float4 -> 2 sync loads (relu + guard)
#pragma unroll
        for (int it = 0; it < 2; ++it) {
            int idx = tid + it * 256;
            int r   = idx >> 3;
            int c4  = (idx & 7) << 2;
            float4 v = make_float4(0.f, 0.f, 0.f, 0.f);
            int gr = rowBase + r;
            if (gr < M) v = *(const float4*)&A[(size_t)gr * K + (k0 + c4)];
            if (reluIn) {
                v.x = fmaxf(v.x, 0.f); v.y = fmaxf(v.y, 0.f);
                v.z = fmaxf(v.z, 0.f); v.w = fmaxf(v.w, 0.f);
            }
            *(float4*)&As[r][c4] = v;
        }
        waitAsync();
        __syncthreads();

#pragma unroll
        for (int kk = 0; kk < 8; ++kk) {
            const int kb = kk * 4 + half * 2; // A lanes0-15 K={0,1}, lanes16-31 K={2,3}
            v2f a = *(const v2f*)&As[wr + lm][kb];
            v2f b0 = { Bs[kb][wc + lm],      Bs[kb + 1][wc + lm]      };
            v2f b1 = { Bs[kb][wc + lm + 16], Bs[kb + 1][wc + lm + 16] };
            acc0 = __builtin_amdgcn_wmma_f32_16x16x4_f32(
                false, a, false, b0, (short)0, acc0, false, false);
            acc1 = __builtin_amdgcn_wmma_f32_16x16x4_f32(
                false, a, false, b1, (short)0, acc1, false, false);
        }
        __syncthreads();
    }

    // D layout: VGPR i -> row i + 8*half, col = lane&15
#pragma unroll
    for (int i = 0; i < 8; ++i) {
        int r = rowBase + wr + i + half * 8;
        if (r < M) {
            C[(size_t)r * Nc + (colBase + wc + lm)]      = acc0[i];
            C[(size_t)r * Nc + (colBase + wc + lm + 16)] = acc1[i];
        }
    }
}

// ------------------------------------------------- self-loop + bias init (F=256)
__global__ __launch_bounds__(256) void k_init256(const float* __restrict__ t,
                                                 const float* __restrict__ dis,
                                                 const float* __restrict__ bias,
                                                 float* __restrict__ agg, int Nn) {
    int q = blockIdx.x * 256 + threadIdx.x;       // over Nn*64 float4 slots
    if (q >= Nn * 64) return;
    int i = q >> 6;
    int f = (q & 63) << 2;
    float s = dis[i]; s = s * s;                  // self-loop norm = dis^2
    float4 v = *(const float4*)&t[(size_t)i * 256 + f];
    float4 b = *(const float4*)&bias[f];
    float4 o;
    o.x = v.x * s + b.x; o.y = v.y * s + b.y;
    o.z = v.z * s + b.z; o.w = v.w * s + b.w;
    *(float4*)&agg[(size_t)i * 256 + f] = o;
}

// -------------------------------------------------------- edge scatter (F=256)
// one wave32 per edge; each lane handles 8 contiguous features
__global__ __launch_bounds__(256) void k_scatter256(const float* __restrict__ t,
                                                    const int* __restrict__ src,
                                                    const int* __restrict__ dst,
                                                    const float* __restrict__ dis,
                                                    float* __restrict__ agg, int E) {
    int gid  = blockIdx.x * 256 + threadIdx.x;
    int e    = gid >> 5;
    if (e >= E) return;
    int lane = gid & 31;
    int s = src[e], d = dst[e];
    float w = dis[s] * dis[d];
    const float* tp = &t[(size_t)s * 256 + lane * 8];
    float*       ap = &agg[(size_t)d * 256 + lane * 8];
    float4 v0 = *(const float4*)(tp + 0);
    float4 v1 = *(const float4*)(tp + 4);
    atomAddF(ap + 0, v0.x * w); atomAddF(ap + 1, v0.y * w);
    atomAddF(ap + 2, v0.z * w); atomAddF(ap + 3, v0.w * w);
    atomAddF(ap + 4, v1.x * w); atomAddF(ap + 5, v1.y * w);
    atomAddF(ap + 6, v1.z * w); atomAddF(ap + 7, v1.w * w);
}

// ------------------------------------------------------ layer 3 GEMM (OUT = 2)
__global__ __launch_bounds__(256) void k_gemm_out2(const float* __restrict__ Hin,
                                                   const float* __restrict__ W3,
                                                   float* __restrict__ T3, int M) {
    __shared__ float Ws[512];
    int tid = threadIdx.x;
    Ws[tid]       = W3[tid];
    Ws[tid + 256] = W3[tid + 256];
    __syncthreads();
    int i = blockIdx.x * 256 + tid;
    if (i >= M) return;
    float a0 = 0.f, a1 = 0.f;
    const float4* row = (const float4*)&Hin[(size_t)i * 256];
#pragma unroll 4
    for (int q = 0; q < 64; ++q) {
        float4 h = row[q];
        h.x = fmaxf(h.x, 0.f); h.y = fmaxf(h.y, 0.f);
        h.z = fmaxf(h.z, 0.f); h.w = fmaxf(h.w, 0.f);
        int k = q << 2;
        a0 += h.x * Ws[(k + 0) * 2] + h.y * Ws[(k + 1) * 2] +
              h.z * Ws[(k + 2) * 2] + h.w * Ws[(k + 3) * 2];
        a1 += h.x * Ws[(k + 0) * 2 + 1] + h.y * Ws[(k + 1) * 2 + 1] +
              h.z * Ws[(k + 2) * 2 + 1] + h.w * Ws[(k + 3) * 2 + 1];
    }
    T3[(size_t)i * 2 + 0] = a0;
    T3[(size_t)i * 2 + 1] = a1;
}

__global__ __launch_bounds__(256) void k_init2(const float* __restrict__ t3,
                                               const float* __restrict__ dis,
                                               const float* __restrict__ b3,
                                               float* __restrict__ agg3, int Nn) {
    int i = blockIdx.x * 256 + threadIdx.x;
    if (i >= Nn) return;
    float s = dis[i]; s = s * s;
    agg3[2 * i + 0] = t3[2 * i + 0] * s + b3[0];
    agg3[2 * i + 1] = t3[2 * i + 1] * s + b3[1];
}

__global__ __launch_bounds__(256) void k_scatter2(const float* __restrict__ t3,
                                                  const int* __restrict__ src,
                                                  const int* __restrict__ dst,
                                                  const float* __restrict__ dis,
                                                  float* __restrict__ agg3, int E) {
    int e = blockIdx.x * 256 + threadIdx.x;
    if (e >= E) return;
    int s = src[e], d = dst[e];
    float w = dis[s] * dis[d];
    atomAddF(&agg3[2 * d + 0], t3[2 * s + 0] * w);
    atomAddF(&agg3[2 * d + 1], t3[2 * s + 1] * w);
}

__global__ __launch_bounds__(256) void k_logsoftmax(const float* __restrict__ agg3,
                                                    float* __restrict__ out, int Nn) {
    int i = blockIdx.x * 256 + threadIdx.x;
    if (i >= Nn) return;
    float a = agg3[2 * i + 0];
    float b = agg3[2 * i + 1];
    float m = fmaxf(a, b);
    float lse = m + logf(expf(a - m) + expf(b - m));
    out[2 * i + 0] = a - lse;
    out[2 * i + 1] = b - lse;
}

// --------------------------------------------------------------------- launch
extern "C" void kernel_launch(void* const* d_in, const int* in_sizes, int n_in,
                              void* d_out, int out_size, void* d_ws, size_t ws_size,
                              hipStream_t stream) {
    const float* x  = (const float*)d_in[0];
    const int*   ei = (const int*)d_in[1];
    const float* W1 = (const float*)d_in[2];
    const float* b1 = (const float*)d_in[3];
    const float* W2 = (const float*)d_in[4];
    const float* b2 = (const float*)d_in[5];
    const float* W3 = (const float*)d_in[6];
    const float* b3 = (const float*)d_in[7];

    const int N = in_sizes[0] / 128;
    const int E = in_sizes[1] / 2;
    const int* src = ei;          // edge_index[0]
    const int* dst = ei + E;      // edge_index[1]

    // workspace carve-up (floats), 64-float aligned chunks
    float* ws  = (float*)d_ws;
    size_t off = 0;
    auto carve = [&](size_t n) { float* p = ws + off; off += (n + 63) & ~(size_t)63; return p; };
    float* dis  = carve((size_t)N);            // deg accumulator, then rsqrt norm
    float* t    = carve((size_t)N * 256);      // h @ W
    float* agg  = carve((size_t)N * 256);      // aggregated (pre-relu)
    float* t3   = carve((size_t)N * 2);
    float* agg3 = carve((size_t)N * 2);

    const dim3 blk(256);
    (void)hipMemsetAsync(dis, 0, (size_t)N * sizeof(float), stream);

    k_degree<<<(E + 255) / 256, blk, 0, stream>>>(dst, dis, E);
    k_rsqrt <<<(N + 255) / 256, blk, 0, stream>>>(dis, N);

    const dim3 gGemm((N + 63) / 64, 256 / 64);

    // layer 1: t = x @ W1 ; agg = A_norm t + b1
    k_gemm_wmma<<<gGemm, blk, 0, stream>>>(x, W1, t, N, 128, 256, 0);
    k_init256  <<<(N * 64 + 255) / 256, blk, 0, stream>>>(t, dis, b1, agg, N);
    k_scatter256<<<(E * 32 + 255) / 256, blk, 0, stream>>>(t, src, dst, dis, agg, E);

    // layer 2: t = relu(agg) @ W2 ; agg = A_norm t + b2
    k_gemm_wmma<<<gGemm, blk, 0, stream>>>(agg, W2, t, N, 256, 256, 1);
    k_init256  <<<(N * 64 + 255) / 256, blk, 0, stream>>>(t, dis, b2, agg, N);
    k_scatter256<<<(E * 32 + 255) / 256, blk, 0, stream>>>(t, src, dst, dis, agg, E);

    // layer 3 (OUT=2): t3 = relu(agg) @ W3 ; agg3 = A_norm t3 + b3 ; log_softmax
    k_gemm_out2<<<(N + 255) / 256, blk, 0, stream>>>(agg, W3, t3, N);
    k_init2    <<<(N + 255) / 256, blk, 0, stream>>>(t3, dis, b3, agg3, N);
    k_scatter2 <<<(E + 255) / 256, blk, 0, stream>>>(t3, src, dst, dis, agg3, E);
    k_logsoftmax<<<(N + 255) / 256, blk, 0, stream>>>(agg3, (float*)d_out, N);
}